// ChannelPair2D_8375186227508
// MI455X (gfx1250) — compile-verified
//
#include <hip/hip_runtime.h>
#include <cstdint>

// -------- CDNA5 feature detection (compile-safe on either toolchain) --------
#ifndef __has_builtin
#define __has_builtin(x) 0
#endif

#if __has_builtin(__builtin_amdgcn_global_load_async_to_lds_b128)
#define USE_ASYNC_LDS 1
#else
#define USE_ASYNC_LDS 0
#endif

#if __has_builtin(__builtin_amdgcn_s_wait_asynccnt)
#define WAIT_ASYNC() __builtin_amdgcn_s_wait_asynccnt(0)
#else
#define WAIT_ASYNC() asm volatile("s_wait_asynccnt 0" ::: "memory")
#endif

typedef float v4f __attribute__((ext_vector_type(4)));
typedef int   v4i __attribute__((ext_vector_type(4)));
// The async builtin wants pointer-to-128-bit-vector in AS1 (global) / AS3 (LDS).
typedef __attribute__((address_space(1))) v4i as1_v4i;
typedef __attribute__((address_space(3))) v4i as3_v4i;

constexpr int kC       = 64;    // channels
constexpr int kK       = 2016;  // C*(C-1)/2 output pairs
constexpr int kK4      = 504;   // kK / 4 (float4 stores per pixel)
constexpr int kPix     = 32;    // pixels per workgroup (8 KB LDS stage)
constexpr int kThreads = 256;   // 8 waves (wave32)

// Map flat pair index k -> (i, j), i<j, row-major triu order.
// offset(i) = 63*i - i*(i-1)/2 ; boundaries 16129-8k = (127-2i)^2 are exact
// perfect squares, so fp32 sqrt is exact there; integer fix-up guarantees
// correctness everywhere else.
__device__ __forceinline__ void pair_of(int k, int& io, int& jo) {
  float s = __builtin_sqrtf((float)(16129 - 8 * k));
  int i = (int)((127.0f - s) * 0.5f);
  i = (i < 0) ? 0 : (i > 62 ? 62 : i);
  while (63 * (i + 1) - ((i + 1) * i) / 2 <= k) ++i;
  while (63 * i - (i * (i - 1)) / 2 > k) --i;
  io = i;
  jo = k - (63 * i - (i * (i - 1)) / 2) + i + 1;
}

__global__ __launch_bounds__(kThreads) void channel_pair_kernel(
    const float* __restrict__ in, float* __restrict__ out) {
  __shared__ float xs[kPix * kC];  // 8 KB: 32 pixels x 64 channels

  const int t = threadIdx.x;
  const long long pix0 = (long long)blockIdx.x * kPix;

  // ---- Stage input slice (contiguous 8 KB) into LDS -----------------------
  // kPix*kC = 2048 floats = 512 x 16B -> two async b128 copies per thread.
  const v4f* __restrict__ in4 = (const v4f*)(in + pix0 * kC);
  v4f* xs4 = (v4f*)xs;
#if USE_ASYNC_LDS
  // CDNA5 async global->LDS copies (ASYNCcnt-tracked, bypass VGPRs).
  __builtin_amdgcn_global_load_async_to_lds_b128(
      (as1_v4i*)(uintptr_t)(in4 + t),
      (as3_v4i*)(uint32_t)(uintptr_t)(xs4 + t), 0, 0);
  __builtin_amdgcn_global_load_async_to_lds_b128(
      (as1_v4i*)(uintptr_t)(in4 + 256 + t),
      (as3_v4i*)(uint32_t)(uintptr_t)(xs4 + 256 + t), 0, 0);
  WAIT_ASYNC();
#else
  xs4[t] = in4[t];
  xs4[256 + t] = in4[256 + t];
#endif
  __syncthreads();

  // ---- Each thread owns k-quads [4t..4t+3] and [4(t+256)..] once ----------
  int ia[8], ja[8];
#pragma unroll
  for (int u = 0; u < 4; ++u) pair_of(4 * t + u, ia[u], ja[u]);
  const bool has2 = (t + 256) < kK4;  // t < 248
  if (has2) {
#pragma unroll
    for (int u = 0; u < 4; ++u) pair_of(4 * (256 + t) + u, ia[4 + u], ja[4 + u]);
  }

  // ---- Stream 32 pixels: LDS broadcast reads -> NT coalesced b128 stores --
  v4f* __restrict__ out4 = (v4f*)out;
#pragma unroll 4
  for (int p = 0; p < kPix; ++p) {
    const float* xp = xs + p * kC;
    const long long ob4 = (pix0 + p) * kK4;  // 16B-aligned (2016*4 % 16 == 0)

    v4f r0;
    r0.x = xp[ia[0]] * xp[ja[0]];
    r0.y = xp[ia[1]] * xp[ja[1]];
    r0.z = xp[ia[2]] * xp[ja[2]];
    r0.w = xp[ia[3]] * xp[ja[3]];
    __builtin_nontemporal_store(r0, out4 + ob4 + t);

    if (has2) {
      v4f r1;
      r1.x = xp[ia[4]] * xp[ja[4]];
      r1.y = xp[ia[5]] * xp[ja[5]];
      r1.z = xp[ia[6]] * xp[ja[6]];
      r1.w = xp[ia[7]] * xp[ja[7]];
      __builtin_nontemporal_store(r1, out4 + ob4 + 256 + t);
    }
  }
}

extern "C" void kernel_launch(void* const* d_in, const int* in_sizes, int n_in,
                              void* d_out, int out_size, void* d_ws, size_t ws_size,
                              hipStream_t stream) {
  (void)n_in; (void)out_size; (void)d_ws; (void)ws_size;
  const float* in = (const float*)d_in[0];
  float* out = (float*)d_out;
  const int pixels = in_sizes[0] / kC;    // B*H*W = 65536
  const int blocks = pixels / kPix;       // 2048
  channel_pair_kernel<<<blocks, kThreads, 0, stream>>>(in, out);
}